// Model_73572789780991
// MI455X (gfx1250) — compile-verified
//
#include <hip/hip_runtime.h>
#include <math.h>

// Problem dims (fixed by reference)
#define B_ 16
#define L_ 2048
#define M_ 32000
#define D_ 512

typedef __attribute__((ext_vector_type(2))) float v2f;
typedef __attribute__((ext_vector_type(8))) float v8f;

// Workspace layout (float offsets). Total ~2.4 MB.
#define OFF_EMB_INV 0            // 32001 (padded to 32004)
#define OFF_POS_INV 32004        // 2048
#define OFF_Q       34052        // 512   renorm'd mask-token embedding
#define OFF_POSQ    34564        // 16*512 renorm'd pos rows at mask_idx
#define OFF_SCORES  42756        // 16*2048 scores -> probabilities (in place)
#define OFF_SELA    75524        // 16*512 sel_output (A matrix of final GEMM)
#define OFF_IP      83716        // 16*32000 logits
#define OFF_TERMS   595716       // 16
#define OFF_TGT     595732       // 16 ints (reinterpreted)

__device__ __forceinline__ float wred(float v) {
#pragma unroll
  for (int o = 16; o > 0; o >>= 1) v += __shfl_xor(v, o, 32);
  return v;
}

// ---------------------------------------------------------------------------
// 1) Per-row inverse norms: inv = 1/max(||row||,1). One wave32 per row.
// ---------------------------------------------------------------------------
__global__ void k_norms(const float* __restrict__ emb,
                        const float* __restrict__ pos,
                        float* __restrict__ ws) {
  int wv   = (blockIdx.x * blockDim.x + threadIdx.x) >> 5;
  int lane = threadIdx.x & 31;
  const float* src;
  float* dst;
  if (wv < M_ + 1) {
    src = emb + (size_t)wv * D_;
    dst = ws + OFF_EMB_INV + wv;
  } else {
    int r = wv - (M_ + 1);
    if (r >= L_) return;  // whole wave exits together (wv uniform per wave)
    src = pos + (size_t)r * D_;
    dst = ws + OFF_POS_INV + r;
  }
  float ss = 0.f;
#pragma unroll
  for (int j = 0; j < D_ / 32; ++j) {
    float v = src[lane + 32 * j];
    ss += v * v;
  }
  ss = wred(ss);
  if (lane == 0) *dst = 1.0f / fmaxf(sqrtf(ss), 1.0f);
}

// ---------------------------------------------------------------------------
// 2) q = ewn[M], posq[b] = pwn[mask_idx[b]], target[b] = x[b, mask_idx[b]]
// ---------------------------------------------------------------------------
__global__ void k_prep(const int* __restrict__ x, const int* __restrict__ mask_idx,
                       const float* __restrict__ emb, const float* __restrict__ pos,
                       float* __restrict__ ws) {
  int tid = threadIdx.x;  // 512 threads
  float invM = ws[OFF_EMB_INV + M_];
  ws[OFF_Q + tid] = emb[(size_t)M_ * D_ + tid] * invM;
  for (int b = 0; b < B_; ++b) {
    int mi = mask_idx[b];
    ws[OFF_POSQ + b * D_ + tid] = pos[(size_t)mi * D_ + tid] * ws[OFF_POS_INV + mi];
  }
  if (tid < B_) {
    int mi = mask_idx[tid];
    ((int*)ws)[OFF_TGT + tid] = x[tid * L_ + mi];
  }
}

// ---------------------------------------------------------------------------
// 3) Scores: s[b,m] = (q.c[b,m] + posq[b].pwn[m]) / sqrt(2D).  One wave/(b,m).
// ---------------------------------------------------------------------------
__global__ void k_scores(const int* __restrict__ x, const int* __restrict__ mask_idx,
                         const float* __restrict__ emb, const float* __restrict__ pos,
                         float* __restrict__ ws) {
  int wv   = (blockIdx.x * blockDim.x + threadIdx.x) >> 5;
  int lane = threadIdx.x & 31;
  int b = wv >> 11;       // / 2048
  int m = wv & (L_ - 1);
  int tok = (m == mask_idx[b]) ? M_ : x[b * L_ + m];
  const float* erow = emb + (size_t)tok * D_;
  const float* prow = pos + (size_t)m * D_;
  const float* q  = ws + OFF_Q;
  const float* pq = ws + OFF_POSQ + b * D_;
  float d1 = 0.f, d2 = 0.f;
#pragma unroll
  for (int j = 0; j < D_ / 32; ++j) {
    int d = lane + 32 * j;
    d1 += q[d] * erow[d];
    d2 += pq[d] * prow[d];
  }
  d1 = wred(d1);
  d2 = wred(d2);
  if (lane == 0) {
    float s = (d1 * ws[OFF_EMB_INV + tok] + d2 * ws[OFF_POS_INV + m]) * (1.0f / 32.0f);
    ws[OFF_SCORES + b * L_ + m] = s;
  }
}

// ---------------------------------------------------------------------------
// 4) Softmax over m (2048) per batch, in place.
// ---------------------------------------------------------------------------
__global__ void k_softmax(float* __restrict__ ws) {
  __shared__ float red[256];
  int b = blockIdx.x, tid = threadIdx.x;
  float* s = ws + OFF_SCORES + b * L_;
  float mx = -1e30f;
  for (int i = tid; i < L_; i += 256) mx = fmaxf(mx, s[i]);
  red[tid] = mx;
  __syncthreads();
  for (int o = 128; o > 0; o >>= 1) {
    if (tid < o) red[tid] = fmaxf(red[tid], red[tid + o]);
    __syncthreads();
  }
  mx = red[0];
  __syncthreads();
  float sum = 0.f;
  for (int i = tid; i < L_; i += 256) sum += expf(s[i] - mx);
  red[tid] = sum;
  __syncthreads();
  for (int o = 128; o > 0; o >>= 1) {
    if (tid < o) red[tid] += red[tid + o];
    __syncthreads();
  }
  float inv = 1.0f / red[0];
  for (int i = tid; i < L_; i += 256) s[i] = expf(s[i] - mx) * inv;
}

// ---------------------------------------------------------------------------
// 5) sel_output[b] = sum_m p[b,m] * content[b,m].  One block per batch,
//    8 waves each own 256 m's, register accumulators, LDS atomic combine.
// ---------------------------------------------------------------------------
__global__ void k_selout(const int* __restrict__ x, const int* __restrict__ mask_idx,
                         const float* __restrict__ emb, float* __restrict__ ws,
                         float* __restrict__ out) {
  __shared__ float acc[D_];
  int b = blockIdx.x, tid = threadIdx.x;
  int lane = tid & 31, wv = tid >> 5;
  for (int i = tid; i < D_; i += 256) acc[i] = 0.f;
  __syncthreads();
  int mi = mask_idx[b];
  float r[16];
#pragma unroll
  for (int j = 0; j < 16; ++j) r[j] = 0.f;
  for (int m = wv * 256; m < wv * 256 + 256; ++m) {
    int tok = (m == mi) ? M_ : x[b * L_ + m];
    float pm = ws[OFF_SCORES + b * L_ + m] * ws[OFF_EMB_INV + tok];
    const float* erow = emb + (size_t)tok * D_;
#pragma unroll
    for (int j = 0; j < 16; ++j) r[j] += pm * erow[lane + 32 * j];
  }
#pragma unroll
  for (int j = 0; j < 16; ++j) atomicAdd(&acc[lane + 32 * j], r[j]);
  __syncthreads();
  for (int i = tid; i < D_; i += 256) {
    float v = acc[i];
    ws[OFF_SELA + b * D_ + i] = v;
    out[1 + b * D_ + i] = v;  // second return value of the reference
  }
}

// ---------------------------------------------------------------------------
// 6) Logits GEMM with WMMA: ip[16 x 32000] = selA[16x512] x ewn[:32000]^T.
//    V_WMMA_F32_16X16X4_F32, M=16 (batch), N tile = 16 vocab rows per wave,
//    K=512 in 128 steps of 4. Row renorm applied as post-scale per N column.
// ---------------------------------------------------------------------------
__global__ void __launch_bounds__(256) k_logits(const float* __restrict__ emb,
                                                float* __restrict__ ws) {
  __shared__ float sA[16 * 514];  // stride 514 spreads LDS banks for b64 reads
  int tid = threadIdx.x;
  for (int i = tid; i < B_ * D_; i += 256)
    sA[(i >> 9) * 514 + (i & 511)] = ws[OFF_SELA + i];
  __syncthreads();

  int wv = tid >> 5, lane = tid & 31;
  int tileN = blockIdx.x * 8 + wv;      // 2000 tiles total
  int col   = lane & 15;                 // N column within tile / A row (M)
  int khalf = lane >> 4;                 // K phase: lanes 16-31 hold K+2,K+3
  int nrow  = tileN * 16 + col;          // vocab row for this lane (< 32000)
  const float* pB = emb + (size_t)nrow * D_;
  const float* pA = sA + col * 514 + 2 * khalf;

  v8f c = {};  // C starts at zero
#pragma unroll 4
  for (int k0 = 0; k0 < D_; k0 += 4) {
    if ((k0 & 63) == 0) __builtin_prefetch(pB + k0 + 128, 0, 1);
    v2f a  = *(const v2f*)(pA + k0);            // A[col, k0+2*khalf .. +1]
    v2f bb = *(const v2f*)(pB + k0 + 2 * khalf);// B[k,col] = ew[nrow, k..k+1]
    c = __builtin_amdgcn_wmma_f32_16x16x4_f32(
        /*neg_a=*/false, a, /*neg_b=*/false, bb,
        /*c_mod=*/(short)0, c, /*reuse_a=*/false, /*reuse_b=*/false);
  }

  float invn = ws[OFF_EMB_INV + nrow];   // renorm factor, constant per lane
  float* ip = ws + OFF_IP;
#pragma unroll
  for (int g = 0; g < 8; ++g) {
    int rowM = g + 8 * khalf;            // C/D layout: lanes 16-31 hold M+8
    ip[(size_t)rowM * M_ + tileN * 16 + col] = c[g] * invn;
  }
}

// ---------------------------------------------------------------------------
// 7) log-softmax over vocab per batch; term_b = logit[target] - max - log(sum)
// ---------------------------------------------------------------------------
__global__ void k_lsm(float* __restrict__ ws) {
  __shared__ float red[512];
  int b = blockIdx.x, tid = threadIdx.x;
  const float* ip = ws + OFF_IP + (size_t)b * M_;
  float mx = -1e30f;
  for (int i = tid; i < M_; i += 512) mx = fmaxf(mx, ip[i]);
  red[tid] = mx;
  __syncthreads();
  for (int o = 256; o > 0; o >>= 1) {
    if (tid < o) red[tid] = fmaxf(red[tid], red[tid + o]);
    __syncthreads();
  }
  mx = red[0];
  __syncthreads();
  float sum = 0.f;
  for (int i = tid; i < M_; i += 512) sum += expf(ip[i] - mx);
  red[tid] = sum;
  __syncthreads();
  for (int o = 256; o > 0; o >>= 1) {
    if (tid < o) red[tid] += red[tid + o];
    __syncthreads();
  }
  if (tid == 0) {
    int tgt = ((const int*)ws)[OFF_TGT + b];
    ws[OFF_TERMS + b] = ip[tgt] - mx - logf(red[0]);
  }
}

__global__ void k_loss(const float* __restrict__ ws, float* __restrict__ out) {
  int lane = threadIdx.x;
  float v = (lane < B_) ? ws[OFF_TERMS + lane] : 0.f;
  v = wred(v);
  if (lane == 0) out[0] = -v / (float)B_;
}

// ---------------------------------------------------------------------------
extern "C" void kernel_launch(void* const* d_in, const int* in_sizes, int n_in,
                              void* d_out, int out_size, void* d_ws, size_t ws_size,
                              hipStream_t stream) {
  (void)in_sizes; (void)n_in; (void)out_size; (void)ws_size;
  const int*   x        = (const int*)d_in[0];
  const int*   mask_idx = (const int*)d_in[1];
  const float* emb      = (const float*)d_in[2];
  const float* pos      = (const float*)d_in[3];
  float* ws  = (float*)d_ws;
  float* out = (float*)d_out;

  int nwaves = (M_ + 1) + L_;                        // 34049 rows
  k_norms  <<<(nwaves + 7) / 8, 256, 0, stream>>>(emb, pos, ws);
  k_prep   <<<1, 512, 0, stream>>>(x, mask_idx, emb, pos, ws);
  k_scores <<<(B_ * L_) / 8, 256, 0, stream>>>(x, mask_idx, emb, pos, ws);
  k_softmax<<<B_, 256, 0, stream>>>(ws);
  k_selout <<<B_, 256, 0, stream>>>(x, mask_idx, emb, ws, out);
  k_logits <<<(M_ / 16) / 8, 256, 0, stream>>>(emb, ws);
  k_lsm    <<<B_, 512, 0, stream>>>(ws);
  k_loss   <<<1, 32, 0, stream>>>(ws, out);
}